// YoloPostProcess_16733192585467
// MI455X (gfx1250) — compile-verified
//
#include <hip/hip_runtime.h>
#include <hip/hip_bf16.h>
#include <stdint.h>

// ---------------------------------------------------------------------------
// YOLO post-process for MI455X (gfx1250).
//  Kernel 1 (decode): fused sigmoid/decode/class-max/threshold. Memory-bound:
//    137MB in @ 23.3TB/s ~ 6us floor. Double-buffered Tensor Data Mover
//    pipeline: each block issues TWO tensor_load_to_lds DMAs (64 cells x 85
//    channels each), computes tile0 under s_wait_tensorcnt(1) while tile1's
//    DMA is in flight (TENSORcnt completes in-order per wave), then drains
//    with s_wait_tensorcnt(0). Writes compact 24B/box records.
//  Kernel 2 (NMS): 16 blocks (1/image) x 512 threads. 50 scores/thread live
//    in VGPRs for all 300 greedy iterations; argmax via LDS tree; boxes are
//    L2-resident (403KB/image) and prefetched with global_prefetch_b8.
// Workspace layout (needs ~9.7MB):
//   [0)                 float4 obox[16][25200]   (class-offset xyxy)
//   [6451200)           float  score[16][25200]  (thresholded)
//   [8064000)           float  cls[16][25200]
// ---------------------------------------------------------------------------

#define NA        3
#define NC        85
#define NCLS      80
#define BS        16
#define NBOX      25200
#define MAXDET    300
#define CONF_T    0.2f
#define IOU_T     0.6f
#define MAX_WH    4096.0f
#define TILE      64                             // cells per TDM tile
#define PAIR      (2 * TILE)                     // cells per block
#define NMS_T     512
#define NMS_K     ((NBOX + NMS_T - 1) / NMS_T)   // 50

typedef unsigned int u32x4 __attribute__((ext_vector_type(4)));
typedef int          i32x4 __attribute__((ext_vector_type(4)));
typedef int          i32x8 __attribute__((ext_vector_type(8)));

#if defined(__has_builtin)
#  if __has_builtin(__builtin_amdgcn_tensor_load_to_lds) && \
      __has_builtin(__builtin_amdgcn_s_wait_tensorcnt)
#    define USE_TDM 1
#  endif
#endif

__device__ __forceinline__ float sigmoidf_(float x) {
  return 1.0f / (1.0f + __expf(-x));
}

#if USE_TDM
// Issue one TDM 2D-tile DMA: tw cells (contiguous, dim0) x 85 channels
// (dim1, stride HW elements) from gsrc into LDS byte offset ldsOff.
__device__ __forceinline__ void tdm_tile_load(const float* gsrc,
                                              unsigned ldsOff, int tw,
                                              int HW) {
  unsigned long long ga = (unsigned long long)(uintptr_t)gsrc;
  u32x4 g0;
  g0.x = 1u;                                    // count=1, user descriptor
  g0.y = ldsOff;                                // lds_addr (bytes)
  g0.z = (unsigned)(ga & 0xFFFFFFFFull);        // global_addr[31:0]
  g0.w = (unsigned)((ga >> 32) & 0x01FFFFFFull) // global_addr[56:32]
       | 0x80000000u;                           // type=2 ("image")
  const unsigned d0 = 0x40000000u;              // huge tensor_dim0 (no OOB)
  const unsigned d1 = 0x00100000u;              // huge tensor_dim1 (no OOB)
  i32x8 g1;
  g1.s0 = 0x00020000;                                   // data_size=4B
  g1.s1 = (int)((d0 & 0xFFFFu) << 16);                  // tensor_dim0 lo
  g1.s2 = (int)(((d0 >> 16) & 0xFFFFu) |
                ((d1 & 0xFFFFu) << 16));                // dim0 hi | dim1 lo
  g1.s3 = (int)(((d1 >> 16) & 0xFFFFu) |
                ((unsigned)tw << 16));                  // dim1 hi | tile_dim0
  g1.s4 = NC;                                           // tile_dim1 = 85
  g1.s5 = (int)HW;                                      // dim0_stride lo32
  g1.s6 = 0;
  g1.s7 = 0;
  i32x4 gz = {0, 0, 0, 0};
#if __clang_major__ >= 23
  i32x8 gz8 = {0, 0, 0, 0, 0, 0, 0, 0};
  __builtin_amdgcn_tensor_load_to_lds(g0, g1, gz, gz, gz8, 0);
#else
  __builtin_amdgcn_tensor_load_to_lds(g0, g1, gz, gz, 0);
#endif
}
#endif

// ---------------------------------------------------------------------------
// Decode: grid = (ceil(HW/128), NA, BS), block = 64 threads (2 waves).
// Input layout: pred[b][a*85+c][y][x]; block stages 2 tiles of 64 cells.
// ---------------------------------------------------------------------------
__global__ __launch_bounds__(TILE) void yolo_decode_kernel(
    const float* __restrict__ pred, const float* __restrict__ anc,
    const int* __restrict__ imgsz, int H, int W, int levelOff,
    float4* __restrict__ oboxes, float* __restrict__ oscores,
    float* __restrict__ ocls)
{
  __shared__ float s_tile[2 * TILE * NC];   // 2 x 21.75KB ping-pong buffers
  const int tx    = threadIdx.x;
  const int a     = blockIdx.y;
  const int b     = blockIdx.z;
  const int HW    = H * W;
  const int cell0 = blockIdx.x * PAIR;      // tile0 start
  const int cell1 = cell0 + TILE;           // tile1 start
  int tw0 = HW - cell0; tw0 = (tw0 < TILE) ? tw0 : TILE;
  int tw1 = HW - cell1; tw1 = (tw1 < TILE) ? tw1 : TILE;
  const bool two  = (tw1 > 0);
  const long long chbase = ((long long)(b * (NA * NC) + a * NC)) * HW;
  const float strd = (float)(*imgsz) / (float)H;   // stride_w == stride_h
  const float aw   = anc[a * 2 + 0];
  const float ah   = anc[a * 2 + 1];

  // Decode one staged tile from LDS buffer sbuf (row pitch twL).
  auto process = [&](const float* sbuf, int twL, int cellBase) {
    if (tx >= twL) return;
    const int cell = cellBase + tx;
    const int gx   = cell % W;
    const int gy   = cell / W;

    const float s0  = sigmoidf_(sbuf[0 * twL + tx]);
    const float s1  = sigmoidf_(sbuf[1 * twL + tx]);
    const float s2  = sigmoidf_(sbuf[2 * twL + tx]);
    const float s3  = sigmoidf_(sbuf[3 * twL + tx]);
    const float obj = sigmoidf_(sbuf[4 * twL + tx]);

    // argmax over class logits == argmax over sigmoid(logit)*obj (obj >= 0).
    float m = sbuf[5 * twL + tx];
    int   mi = 0;
    for (int c = 1; c < NCLS; ++c) {
      float v = sbuf[(5 + c) * twL + tx];
      if (v > m) { m = v; mi = c; }   // strict > keeps first max (jnp.argmax)
    }
    const float score = obj * sigmoidf_(m);
    const float sc    = (score > CONF_T) ? score : 0.0f;

    const float x = (3.0f * s0 - 1.0f + (float)gx) * strd;
    const float y = (3.0f * s1 - 1.0f + (float)gy) * strd;
    const float w = (2.0f * s2) * (2.0f * s2) * aw;  // (2*sig)^2 * anchor_px
    const float h = (2.0f * s3) * (2.0f * s3) * ah;

    const float clsf = (float)mi;
    const float off  = clsf * MAX_WH;                // class-offset trick
    float4 ob = make_float4(x - 0.5f * w + off, y - 0.5f * h + off,
                            x + 0.5f * w + off, y + 0.5f * h + off);

    const int bi = b * NBOX + levelOff + a * HW + cell;
    oboxes[bi]  = ob;
    oscores[bi] = sc;
    ocls[bi]    = clsf;
  };

#if USE_TDM
  // Wave 0 issues both DMAs back-to-back; TENSORcnt completes in-order, so
  // s_wait_tensorcnt(1) guarantees tile0 landed while tile1 is in flight.
  if (tx < 32) {
    tdm_tile_load(pred + chbase + cell0, 0u, tw0, HW);
    if (two) {
      tdm_tile_load(pred + chbase + cell1, (unsigned)(TILE * NC * 4), tw1, HW);
      __builtin_amdgcn_s_wait_tensorcnt(1);
    } else {
      __builtin_amdgcn_s_wait_tensorcnt(0);
    }
  }
  // Keep the compiler from proving s_tile is never stored (TDM writes it).
  {
    void* esc = (void*)s_tile;
    asm volatile("" : "+v"(esc) :: "memory");
  }
  __syncthreads();
  process(s_tile, tw0, cell0);            // overlaps with tile1 DMA
  if (two) {
    if (tx < 32) __builtin_amdgcn_s_wait_tensorcnt(0);
    __syncthreads();
    process(s_tile + TILE * NC, tw1, cell1);
  }
#else
  // Fallback: cooperative coalesced copies (tx contiguous per channel row).
  if (tx < tw0)
    for (int c = 0; c < NC; ++c)
      s_tile[c * tw0 + tx] = pred[chbase + (long long)c * HW + cell0 + tx];
  if (two && tx < tw1)
    for (int c = 0; c < NC; ++c)
      s_tile[TILE * NC + c * tw1 + tx] =
          pred[chbase + (long long)c * HW + cell1 + tx];
  __syncthreads();
  process(s_tile, tw0, cell0);
  if (two) process(s_tile + TILE * NC, tw1, cell1);
#endif
}

// ---------------------------------------------------------------------------
// NMS: grid = 16 (one image per workgroup), block = 512 threads (16 waves).
// Scores live in VGPRs (50/thread, strided ownership idx = k*512 + t).
// ---------------------------------------------------------------------------
__global__ __launch_bounds__(NMS_T) void yolo_nms_kernel(
    const float4* __restrict__ boxes, const float* __restrict__ scores,
    const float* __restrict__ cls, float* __restrict__ out)
{
  __shared__ float s_val[NMS_T];
  __shared__ int   s_idx[NMS_T];
  const int b    = blockIdx.x;
  const int t    = threadIdx.x;
  const int base = b * NBOX;

  float sc[NMS_K];
#pragma unroll
  for (int k = 0; k < NMS_K; ++k) {
    int idx = k * NMS_T + t;
    sc[k] = (idx < NBOX) ? scores[base + idx] : 0.0f;
  }

  for (int it = 0; it < MAXDET; ++it) {
    // 1) thread-local argmax over register scores
    float lv = -1.0f;
    int   li = t;
#pragma unroll
    for (int k = 0; k < NMS_K; ++k) {
      if (sc[k] > lv) { lv = sc[k]; li = k * NMS_T + t; }
    }
    s_val[t] = lv;
    s_idx[t] = li;
    __syncthreads();

    // 2) LDS tree reduce to global argmax (tie -> lowest linear index)
    for (int o = NMS_T / 2; o > 0; o >>= 1) {
      if (t < o) {
        float v2 = s_val[t + o];
        int   i2 = s_idx[t + o];
        if (v2 > s_val[t] || (v2 == s_val[t] && i2 < s_idx[t])) {
          s_val[t] = v2;
          s_idx[t] = i2;
        }
      }
      __syncthreads();
    }
    const float bestv = s_val[0];
    const int   besti = s_idx[0];
    __syncthreads();   // all reads done before next iteration's writes

    const bool valid = bestv > 0.0f;
    float4 bb   = make_float4(0.f, 0.f, 0.f, 0.f);
    float  bcls = 0.0f;

    if (valid) {
      bb   = boxes[base + besti];   // uniform address -> scalarized load
      bcls = cls[base + besti];
      const float a1 = (bb.z - bb.x) * (bb.w - bb.y);
      // 3) suppression over this thread's candidates
#pragma unroll
      for (int k = 0; k < NMS_K; ++k) {
        if (k + 1 < NMS_K)   // gfx1250 global_prefetch_b8 for next tile
          __builtin_prefetch((const void*)(boxes + base + (k + 1) * NMS_T + t),
                             0, 0);
        int idx = k * NMS_T + t;
        if (sc[k] > 0.0f) {
          if (idx == besti) {
            sc[k] = 0.0f;
          } else {
            float4 c4 = boxes[base + idx];
            float lx = fmaxf(bb.x, c4.x), ly = fmaxf(bb.y, c4.y);
            float rx = fminf(bb.z, c4.z), ry = fminf(bb.w, c4.w);
            float iw = fmaxf(rx - lx, 0.0f), ih = fmaxf(ry - ly, 0.0f);
            float inter = iw * ih;
            float a2 = (c4.z - c4.x) * (c4.w - c4.y);
            float iou = inter / (a1 + a2 - inter + 1e-9f);
            if (iou > IOU_T) sc[k] = 0.0f;
          }
        }
      }
    }

    // 4) emit detection row (zeros when invalid, matching reference)
    if (t == 0) {
      float* o6 = out + ((size_t)b * MAXDET + it) * 6;
      if (valid) {
        const float ofs = bcls * MAX_WH;
        o6[0] = bb.x - ofs; o6[1] = bb.y - ofs;
        o6[2] = bb.z - ofs; o6[3] = bb.w - ofs;
        o6[4] = bestv;      o6[5] = bcls;
      } else {
        o6[0] = 0.f; o6[1] = 0.f; o6[2] = 0.f;
        o6[3] = 0.f; o6[4] = 0.f; o6[5] = 0.f;
      }
    }
    __syncthreads();
  }
}

// ---------------------------------------------------------------------------
extern "C" void kernel_launch(void* const* d_in, const int* in_sizes, int n_in,
                              void* d_out, int out_size, void* d_ws,
                              size_t ws_size, hipStream_t stream) {
  (void)in_sizes; (void)n_in; (void)out_size; (void)ws_size;
  const float* p0  = (const float*)d_in[0];   // (16,255,80,80)
  const float* p1  = (const float*)d_in[1];   // (16,255,40,40)
  const float* p2  = (const float*)d_in[2];   // (16,255,20,20)
  const float* anc = (const float*)d_in[3];   // (3,3,2)
  const int*   img = (const int*)d_in[4];     // scalar 640

  float4* wsBoxes  = (float4*)d_ws;
  float*  wsScores = (float*)((char*)d_ws + (size_t)BS * NBOX * sizeof(float4));
  float*  wsCls    = wsScores + (size_t)BS * NBOX;
  float*  out      = (float*)d_out;

  const dim3 blk(TILE);
  // pairs of 64-cell tiles: 6400 -> 50, 1600 -> 13 (last block single tile),
  // 400 -> 4 (last block single 16-cell tile)
  yolo_decode_kernel<<<dim3(50, NA, BS), blk, 0, stream>>>(
      p0, anc + 0, img, 80, 80, 0, wsBoxes, wsScores, wsCls);
  yolo_decode_kernel<<<dim3(13, NA, BS), blk, 0, stream>>>(
      p1, anc + 6, img, 40, 40, 19200, wsBoxes, wsScores, wsCls);
  yolo_decode_kernel<<<dim3(4, NA, BS), blk, 0, stream>>>(
      p2, anc + 12, img, 20, 20, 24000, wsBoxes, wsScores, wsCls);

  yolo_nms_kernel<<<dim3(BS), dim3(NMS_T), 0, stream>>>(
      wsBoxes, wsScores, wsCls, out);
}